// AdaptiveStieltjes_80169859547695
// MI455X (gfx1250) — compile-verified
//
#include <hip/hip_runtime.h>

typedef float v2f __attribute__((ext_vector_type(2)));
typedef float v4f __attribute__((ext_vector_type(4)));
typedef float v8f __attribute__((ext_vector_type(8)));

#define KV   2048   // row length (kv_len)
#define EPW  64     // elements per lane = KV / 32 (wave32)
#define V4W  16     // float4 chunks per lane

// Hardware transcendentals: v_exp_f32 (exp2) and v_log_f32 (log2)
__device__ __forceinline__ float fexp2(float x) { return __builtin_amdgcn_exp2f(x); }
__device__ __forceinline__ float flog2(float x) { return __builtin_amdgcn_logf(x); }

// ---- wave32 horizontal sum + broadcast, entirely on the matrix pipe -----
// WMMA #1: A[L][0]=partial (lanes 0..15), A[L-16][2]=partial (lanes 16..31),
// B = all-ones (4x16)  =>  D[m][n] = partial[m] + partial[m+16], independent
// of n. Summing the 8 D VGPRs per lane leaves Σrows0..7 in every low-half
// lane and Σrows8..15 in every high-half lane.
// WMMA #2: feed that back as A with the same ones-B:
//   D2[m][n] = Σ0..7 + Σ8..15 = full 32-lane total, in every lane of d2[0].
// No ds_bpermute / s_wait_dscnt on the serial bisection critical path.
// Exact f32 FMA math. Requires EXEC == all ones (guaranteed below).
__device__ __forceinline__ float wave_sum_wmma(float partial) {
  v2f ones; ones[0] = 1.0f; ones[1] = 1.0f;
  v8f c = {0.f,0.f,0.f,0.f,0.f,0.f,0.f,0.f};

  v2f a; a[0] = partial; a[1] = 0.0f;
  v8f d = __builtin_amdgcn_wmma_f32_16x16x4_f32(
      false, a, false, ones, (short)0, c, false, false);
  float s = ((d[0] + d[1]) + (d[2] + d[3])) + ((d[4] + d[5]) + (d[6] + d[7]));

  v2f a2; a2[0] = s; a2[1] = 0.0f;
  v8f d2 = __builtin_amdgcn_wmma_f32_16x16x4_f32(
      false, a2, false, ones, (short)0, c, false, false);
  return d2[0];
}

__device__ __forceinline__ float wave_max(float v) {
#pragma unroll
  for (int m = 16; m > 0; m >>= 1) v = fmaxf(v, __shfl_xor(v, m, 32));
  return v;
}

__global__ __launch_bounds__(256)
void AdaptiveStieltjes_kernel(const float* __restrict__ logits,
                              float* __restrict__ out, int rows) {
  const int wave = (int)((blockIdx.x * blockDim.x + threadIdx.x) >> 5);
  const int lane = (int)(threadIdx.x & 31);
  if (wave >= rows) return;  // wave-uniform: exiting waves never reach WMMA

  // ---- load one row (8 KB) into registers, fully coalesced, streaming ----
  const v4f* __restrict__ rp =
      (const v4f*)(logits + (size_t)wave * KV);
  float x[EPW];
#pragma unroll
  for (int j = 0; j < V4W; ++j) {
    v4f t = __builtin_nontemporal_load(rp + lane + 32 * j);
    x[4 * j + 0] = t[0]; x[4 * j + 1] = t[1];
    x[4 * j + 2] = t[2]; x[4 * j + 3] = t[3];
  }

  const float L2E = 1.44269504088896340736f;  // log2(e)
  const float LN2 = 0.69314718055994530942f;  // ln(2)
  const float EPS = 1e-9f;

  // ---- entropy of softmax over raw (unclipped) logits --------------------
  float lmax = -INFINITY;
#pragma unroll
  for (int j = 0; j < EPW; ++j) lmax = fmaxf(lmax, x[j]);
  const float rmax = wave_max(lmax);

  float se = 0.f, sxe = 0.f;
#pragma unroll
  for (int j = 0; j < EPW; ++j) {
    float xi = x[j] - rmax;
    float e  = fexp2(xi * L2E);
    se  += e;
    sxe += xi * e;
  }
  se  = wave_sum_wmma(se);
  sxe = wave_sum_wmma(sxe);
  // entropy = log(S) - (1/S) * sum(xi * exp(xi))
  const float entropy = flog2(se) * LN2 - sxe / se;

  // ---- adaptive exponent q via Horner polynomial -------------------------
  float p = -0.037f;
  p = p * entropy + 0.481f;
  p = p * entropy - 2.3f;
  p = p * entropy + 4.917f;
  p = p * entropy - 1.791f;
  p = fminf(fmaxf(p, 1.0f), 10.0f);
  float q = (entropy > 0.5f) ? p : 1.0f;
  q += EPS;
  const float nq = -q;

  // ---- shifted clipped logits: x_i = clip(x) - max(clip(x)) --------------
  // clamp is monotone, so max(clip(x)) == clip(max(x))
  const float xmax = fminf(fmaxf(rmax, -50.f), 50.f);
#pragma unroll
  for (int j = 0; j < EPW; ++j)
    x[j] = fminf(fmaxf(x[j], -50.f), 50.f) - xmax;

  // ---- 32-step bisection for lambda: sum((lambda - x_i)^-q) = 1 ----------
  float lb = EPS;
  float ub = fexp2(11.0f / q);  // n^(1/q), n = 2048 = 2^11
#pragma unroll 1
  for (int it = 0; it < 32; ++it) {
    const float mid = 0.5f * (lb + ub);
    float ps = 0.f;
#pragma unroll
    for (int j = 0; j < EPW; ++j) {
      float t = fmaxf(mid - x[j], EPS);
      ps += fexp2(nq * flog2(t));  // (mid - x_i)^(-q)
    }
    const float prob_sum = wave_sum_wmma(ps) - 1.0f;  // uniform across wave
    const bool gt = prob_sum > 0.f;
    lb = gt ? mid : lb;
    ub = gt ? ub  : mid;
  }
  const float lam = 0.5f * (lb + ub);

  // ---- output: (clip(lambda - x_i, EPS))^(-q), streaming stores ----------
  v4f* __restrict__ op = (v4f*)(out + (size_t)wave * KV);
#pragma unroll
  for (int j = 0; j < V4W; ++j) {
    v4f o;
#pragma unroll
    for (int k = 0; k < 4; ++k) {
      float t = fmaxf(lam - x[4 * j + k], EPS);
      o[k] = fexp2(nq * flog2(t));
    }
    __builtin_nontemporal_store(o, op + lane + 32 * j);
  }
}

extern "C" void kernel_launch(void* const* d_in, const int* in_sizes, int n_in,
                              void* d_out, int out_size, void* d_ws, size_t ws_size,
                              hipStream_t stream) {
  const float* logits = (const float*)d_in[0];
  float* out = (float*)d_out;
  const int rows = in_sizes[0] / KV;          // 2*16*2048 = 65536 rows
  const int waves_per_block = 256 / 32;       // 8 rows per block
  const int blocks = (rows + waves_per_block - 1) / waves_per_block;
  AdaptiveStieltjes_kernel<<<blocks, 256, 0, stream>>>(logits, out, rows);
}